// CRF_15891378995200
// MI455X (gfx1250) — compile-verified
//
#include <hip/hip_runtime.h>
#include <math.h>

// Problem constants (from reference)
#define BB   64
#define TT   512
#define FF   1024
#define NT   50
#define HH   52          // NT + 2 (13 * 4)
#define HP   64          // padded H (emit row stride)
#define STARTS 50
#define STOPS  51
#define NEGV (-100000000.0f)

typedef float v2f __attribute__((ext_vector_type(2)));
typedef float v4f __attribute__((ext_vector_type(4)));
typedef float v8f __attribute__((ext_vector_type(8)));

// ---------------------------------------------------------------------------
// Kernel 1: emit = features @ W^T + b   (32768 x 52, K=1024), fp32 WMMA
// grid 256 blocks x 256 threads; each wave: 16 rows x 64 cols.
// A/B operands fetched as b128 (both halves load K0..K3 of the lane's row;
// duplicate addresses merge in the memory system), halves select their K-pair.
// ---------------------------------------------------------------------------
__global__ __launch_bounds__(256) void emit_gemm(const float* __restrict__ feat,
                                                 const float* __restrict__ W,
                                                 const float* __restrict__ bias,
                                                 float* __restrict__ emit) {
    const int wave = threadIdx.x >> 5;
    const int lane = threadIdx.x & 31;
    const int half = lane >> 4;      // 0 -> K0/K1 VGPR halves, 1 -> K2/K3
    const int l16  = lane & 15;
    const int row0 = blockIdx.x * 128 + wave * 16;

    const v4f* __restrict__ arow = (const v4f*)(feat + (size_t)(row0 + l16) * FF);

    // B-matrix rows: column h of W^T is row h of W; pad h>=HH with zeros
    const v4f* wrow[4];
    float wmask[4];
    #pragma unroll
    for (int n = 0; n < 4; ++n) {
        int h = n * 16 + l16;
        wmask[n] = (h < HH) ? 1.0f : 0.0f;
        wrow[n]  = (const v4f*)(W + (size_t)((h < HH) ? h : 0) * FF);
    }

    v8f acc[4] = {};

    #pragma unroll 4
    for (int k4 = 0; k4 < FF / 4; ++k4) {
        const v4f q = arow[k4];                       // K0..K3 of this lane's row
        v2f a;
        a.x = half ? q.z : q.x;                       // VGPR0: K0 (lo) / K2 (hi)
        a.y = half ? q.w : q.y;                       // VGPR1: K1 (lo) / K3 (hi)
        #pragma unroll
        for (int n = 0; n < 4; ++n) {
            const v4f p = wrow[n][k4];
            v2f bb;
            bb.x = (half ? p.z : p.x) * wmask[n];
            bb.y = (half ? p.w : p.y) * wmask[n];
            acc[n] = __builtin_amdgcn_wmma_f32_16x16x4_f32(
                false, a, false, bb, (short)0, acc[n], false, false);
        }
    }

    // C/D layout: VGPR r -> M = r (lanes 0-15), M = r + 8 (lanes 16-31); N = l16
    #pragma unroll
    for (int n = 0; n < 4; ++n) {
        const int col = n * 16 + l16;
        const float bv = (col < HH) ? bias[col] : 0.0f;
        #pragma unroll
        for (int r = 0; r < 8; ++r) {
            emit[(size_t)(row0 + r + 8 * half) * HP + col] = acc[n][r] + bv;
        }
    }
}

// ---------------------------------------------------------------------------
// Kernel 2: CRF forward recursion, exp-factored:
//   u_i = sum_j exp(trans[i][j]) * exp(score_j - m);  score_i = emit + m + log(u_i)
// One block per batch element; 64 threads (2 waves); scores live in LDS.
// LDS traffic vectorized as float4 (52 = 13*4); emit/mask loads hoisted to the
// top of each step so global latency hides behind the max/exp phase.
// ---------------------------------------------------------------------------
__global__ __launch_bounds__(64) void crf_scan(const float* __restrict__ emit,
                                               const float* __restrict__ trans,
                                               const float* __restrict__ masks,
                                               float* __restrict__ fwd) {
    const int b   = blockIdx.x;
    const int tid = threadIdx.x;

    __shared__ __align__(16) float sc[HH];
    __shared__ __align__(16) float vv[HH];

    v4f Erow[HH / 4];
    if (tid < HH) {
        #pragma unroll 1
        for (int j4 = 0; j4 < HH / 4; ++j4) {
            v4f e;
            e.x = expf(trans[tid * HH + 4 * j4 + 0]);   // NEG -> exactly 0
            e.y = expf(trans[tid * HH + 4 * j4 + 1]);
            e.z = expf(trans[tid * HH + 4 * j4 + 2]);
            e.w = expf(trans[tid * HH + 4 * j4 + 3]);
            Erow[j4] = e;
        }
        sc[tid] = (tid == STARTS) ? 0.0f : NEGV;
    }
    __syncthreads();

    const float* __restrict__ erow = emit + (size_t)b * TT * HP;
    const float* __restrict__ mrow = masks + (size_t)b * TT;

    for (int t = 0; t < TT; ++t) {
        // issue this step's global loads first; latency hides behind max+exp
        const float e_t = erow[(size_t)t * HP + tid];
        const float mk  = mrow[t];

        // per-batch max over states: float4 LDS reads, pairwise fold
        v4f mm = ((const v4f*)sc)[0];
        #pragma unroll
        for (int j4 = 1; j4 < HH / 4; ++j4) {
            v4f s4 = ((const v4f*)sc)[j4];
            mm.x = fmaxf(mm.x, s4.x); mm.y = fmaxf(mm.y, s4.y);
            mm.z = fmaxf(mm.z, s4.z); mm.w = fmaxf(mm.w, s4.w);
        }
        const float m = fmaxf(fmaxf(mm.x, mm.y), fmaxf(mm.z, mm.w));

        if (tid < HH) vv[tid] = expf(sc[tid] - m);
        __syncthreads();

        if (tid < HH) {
            v4f u4 = {};
            #pragma unroll
            for (int j4 = 0; j4 < HH / 4; ++j4) {
                const v4f x = ((const v4f*)vv)[j4];
                const v4f e = Erow[j4];
                u4.x = fmaf(e.x, x.x, u4.x); u4.y = fmaf(e.y, x.y, u4.y);
                u4.z = fmaf(e.z, x.z, u4.z); u4.w = fmaf(e.w, x.w, u4.w);
            }
            const float u  = (u4.x + u4.y) + (u4.z + u4.w);
            const float ns = e_t + m + logf(u);
            sc[tid] = (mk != 0.0f) ? ns : sc[tid];
        }
        __syncthreads();
    }

    if (tid == 0) {
        float m2 = -INFINITY;
        for (int j = 0; j < HH; ++j) m2 = fmaxf(m2, sc[j] + trans[STOPS * HH + j]);
        float s = 0.0f;
        for (int j = 0; j < HH; ++j) s += expf(sc[j] + trans[STOPS * HH + j] - m2);
        fwd[b] = m2 + logf(s);
    }
}

// ---------------------------------------------------------------------------
// Kernel 3: gold path score per batch
// ---------------------------------------------------------------------------
__global__ __launch_bounds__(256) void crf_gold(const float* __restrict__ emit,
                                                const float* __restrict__ trans,
                                                const float* __restrict__ masks,
                                                const int* __restrict__ tags,
                                                float* __restrict__ gold) {
    const int b   = blockIdx.x;
    const int tid = threadIdx.x;
    __shared__ float red[256];
    __shared__ float redm[256];

    float acc = 0.0f, msum = 0.0f;
    for (int t = tid; t < TT; t += 256) {
        int   tg = tags[b * TT + t];
        int   tp = (t == 0) ? STARTS : tags[b * TT + t - 1];
        float mk = masks[b * TT + t];
        acc  += (trans[tg * HH + tp] + emit[((size_t)b * TT + t) * HP + tg]) * mk;
        msum += mk;
    }
    red[tid] = acc; redm[tid] = msum;
    __syncthreads();
    for (int s = 128; s > 0; s >>= 1) {
        if (tid < s) { red[tid] += red[tid + s]; redm[tid] += redm[tid + s]; }
        __syncthreads();
    }
    if (tid == 0) {
        int last = (int)(redm[0] + 0.5f);
        int lt   = (last == 0) ? STARTS : tags[b * TT + last - 1];
        gold[b]  = red[0] + trans[STOPS * HH + lt];
    }
}

// ---------------------------------------------------------------------------
// Kernel 4: out = mean(fwd - gold)
// ---------------------------------------------------------------------------
__global__ __launch_bounds__(64) void finalize(const float* __restrict__ fwd,
                                               const float* __restrict__ gold,
                                               float* __restrict__ out) {
    const int tid = threadIdx.x;
    __shared__ float r[64];
    r[tid] = fwd[tid] - gold[tid];
    __syncthreads();
    for (int s = 32; s > 0; s >>= 1) {
        if (tid < s) r[tid] += r[tid + s];
        __syncthreads();
    }
    if (tid == 0) out[0] = r[0] * (1.0f / BB);
}

// ---------------------------------------------------------------------------
extern "C" void kernel_launch(void* const* d_in, const int* in_sizes, int n_in,
                              void* d_out, int out_size, void* d_ws, size_t ws_size,
                              hipStream_t stream) {
    const float* features = (const float*)d_in[0];
    const float* W        = (const float*)d_in[1];
    const float* bias     = (const float*)d_in[2];
    const float* trans    = (const float*)d_in[3];
    const float* masks    = (const float*)d_in[4];
    const int*   tags     = (const int*)d_in[5];
    float*       out      = (float*)d_out;

    float* emit = (float*)d_ws;                         // 32768 x 64 fp32 = 8.39 MB
    float* fwd  = emit + (size_t)BB * TT * HP;          // 64 fp32
    float* gold = fwd + 64;                             // 64 fp32

    emit_gemm<<<256, 256, 0, stream>>>(features, W, bias, emit);
    crf_scan <<<BB,  64, 0, stream>>>(emit, trans, masks, fwd);
    crf_gold <<<BB, 256, 0, stream>>>(emit, trans, masks, tags, gold);
    finalize <<<1,   64, 0, stream>>>(fwd, gold, out);
}